// ObjectCondensationLoss_44049184588447
// MI455X (gfx1250) — compile-verified
//
#include <hip/hip_runtime.h>
#include <hip/hip_bf16.h>
#include <float.h>
#include <limits.h>
#include <math.h>

#define B_ 4
#define N_ 4096
#define D_ 32
#define K_ 64

__device__ __constant__ const float c_ATTR_W  = 1.0f;
__device__ __constant__ const float c_REP_W   = 1.5f;
__device__ __constant__ const float c_MARG_W  = 5.0f;
__device__ __constant__ const float c_NONCP_W = 0.5f;
// THRESH=0.5, MARGIN=0.2 -> pos edge 0.7, neg edge 0.3

typedef __attribute__((ext_vector_type(2))) float v2f;
typedef __attribute__((ext_vector_type(8))) float v8f;

__device__ __forceinline__ float clean_f(float v) {
    if (isnan(v)) return 0.0f;
    if (isinf(v)) return copysignf(FLT_MAX, v);
    return v;
}

// ---------------- K0: init workspace accumulators ----------------
__global__ void k_init(float* seg, float* rep, float* acc,
                       int* first, int* hist, int* cnt, int* ncp) {
    int i = blockIdx.x * blockDim.x + threadIdx.x;
    if (i < B_ * K_) { seg[i] = 0.0f; first[i] = INT_MAX; hist[i] = 0; cnt[i] = 0; }
    if (i < B_)      { rep[i] = 0.0f; ncp[i] = 0; }
    if (i < B_ * 4)  { acc[i] = 0.0f; }
}

// ---------------- K1: clean inputs, norms+key, first-cp index, histograms ----------------
// attr[idx] = { |e|^2 , cp ? (float)sid : -1.0f }  -> one b64 load per point in k_rep
__global__ void k_prep(const float* __restrict__ beta,
                       const float* __restrict__ embed,
                       const int* __restrict__ sid_g,
                       const int* __restrict__ cp_g,
                       float* __restrict__ wemb, float2* __restrict__ attr,
                       float* __restrict__ wbeta,
                       int* first, int* hist, int* cnt, int* ncp) {
    int idx = blockIdx.x * blockDim.x + threadIdx.x;
    if (idx >= B_ * N_) return;
    wbeta[idx] = clean_f(beta[idx]);

    const float* e = embed + (size_t)idx * D_;
    float*      we = wemb  + (size_t)idx * D_;
    float s = 0.0f;
#pragma unroll
    for (int d = 0; d < D_; ++d) {
        float v = clean_f(e[d]);
        we[d] = v;
        s += v * v;
    }
    int b = idx / N_, n = idx % N_;
    int sid = sid_g[idx];
    bool cp = (cp_g[idx] == 1);
    attr[idx] = make_float2(s, cp ? (float)sid : -1.0f);

    atomicAdd(&cnt[b * K_ + sid], 1);
    if (cp) {
        atomicMin(&first[b * K_ + sid], n);
        atomicAdd(&hist[b * K_ + sid], 1);
        atomicAdd(&ncp[b], 1);
    }
}

// ---------------- K2: attraction segment sums + beta-loss reductions ----------------
__global__ void __launch_bounds__(256)
k_point(const int* __restrict__ sid_g, const int* __restrict__ cp_g,
        const float* __restrict__ wemb, const float* __restrict__ wbeta,
        const int* __restrict__ first, const int* __restrict__ ncp,
        float* seg, float* acc) {
    int idx = blockIdx.x * blockDim.x + threadIdx.x;   // grid == B_*N_ exactly
    int b = idx / N_;
    int sid = sid_g[idx];
    bool cp = (cp_g[idx] == 1);

    // attraction: distance to first cp point of this instance (anchor idx 0 if none)
    int fi = first[b * K_ + sid];
    if (fi == INT_MAX) fi = 0;
    const float* e = wemb + (size_t)idx * D_;
    const float* a = wemb + ((size_t)b * N_ + fi) * D_;
    float d2 = 0.0f;
#pragma unroll
    for (int d = 0; d < D_; ++d) { float t = e[d] - a[d]; d2 += t * t; }
    d2 = fminf(d2, 50.0f);
    atomicAdd(&seg[b * K_ + sid], d2);

    // beta loss pieces
    float x = wbeta[idx];
    x = fminf(20.0f, fmaxf(-20.0f, x));
    float fncp   = (float)ncp[b];
    float pos_wt = ((float)N_ - fncp) / (fncp + 1e-6f);
    float log_sig  = -log1pf(__expf(-x));
    float log_1ms  = -log1pf(__expf(x));
    float cpf = cp ? 1.0f : 0.0f;
    float nf  = 1.0f - cpf;
    float bce = -(pos_wt * cpf * log_sig + nf * log_1ms);
    float prob = 1.0f / (1.0f + __expf(-x));
    float pm = cp ? fmaxf(0.7f - prob, 0.0f) : 0.0f;
    float nm = cp ? 0.0f : fmaxf(prob - 0.3f, 0.0f);
    float sp = cp ? 0.0f : prob * prob;

    __shared__ float red[256];
    float vals[4] = {bce, pm, nm, sp};
#pragma unroll
    for (int q = 0; q < 4; ++q) {
        red[threadIdx.x] = vals[q];
        __syncthreads();
        for (int s2 = 128; s2 > 0; s2 >>= 1) {
            if ((int)threadIdx.x < s2) red[threadIdx.x] += red[threadIdx.x + s2];
            __syncthreads();
        }
        if (threadIdx.x == 0) atomicAdd(&acc[b * 4 + q], red[0]);
        __syncthreads();
    }
}

// ---------------- K3: repulsion via WMMA f32 Gram tiles (lower triangle only) -------
// One wave32 computes one 16x16 tile of G = E * E^T with 8x V_WMMA_F32_16X16X4_F32.
// Symmetry: off-diagonal tiles weighted 2x. Masking is fully branchless.
__global__ void __launch_bounds__(256)
k_rep(const float* __restrict__ wemb, const float2* __restrict__ attr, float* rep) {
    const int T = N_ / 16;                     // 256 tiles per dim
    const int P = T * (T + 1) / 2;             // 32896 lower-triangle tiles per batch
    int wave = (int)((blockIdx.x * blockDim.x + threadIdx.x) >> 5);
    int lane = (int)(threadIdx.x & 31);
    int b = wave / P;
    int t = wave % P;
    // decode lower-triangle index: rt*(rt+1)/2 <= t < (rt+1)*(rt+2)/2, ct <= rt
    int rt = (int)((sqrt(8.0 * (double)t + 1.0) - 1.0) * 0.5);
    while ((rt + 1) * (rt + 2) / 2 <= t) ++rt;
    while (rt * (rt + 1) / 2 > t) --rt;
    int ct = t - rt * (rt + 1) / 2;
    float scale = (rt == ct) ? 1.0f : 2.0f;

    int half = lane >> 4;                      // 0: lanes 0-15, 1: lanes 16-31
    int lm   = lane & 15;
    int rowBase = rt * 16, colBase = ct * 16;

    // A: 16x4 f32 tile (rows); lane holds M=lm, VGPR0/1 = K pair selected by half.
    // B: 4x16 f32 tile (cols of E^T): same lane->(point, K-pair) pattern.
    const float* erow = wemb + ((size_t)b * N_ + rowBase + lm) * D_;
    const float* ecol = wemb + ((size_t)b * N_ + colBase + lm) * D_;

    v8f c = {0.f, 0.f, 0.f, 0.f, 0.f, 0.f, 0.f, 0.f};
#pragma unroll
    for (int kk = 0; kk < D_ / 4; ++kk) {
        int kb = 4 * kk + 2 * half;
        v2f av = { erow[kb], erow[kb + 1] };
        v2f bv = { ecol[kb], ecol[kb + 1] };
        c = __builtin_amdgcn_wmma_f32_16x16x4_f32(
                /*neg_a=*/false, av, /*neg_b=*/false, bv,
                /*c_mod=*/(short)0, c, /*reuse_a=*/false, /*reuse_b=*/false);
    }

    // C/D layout: VGPR r, lane -> row (r + 8*half), col lm
    float2 aj = attr[b * N_ + colBase + lm];
    float accv = 0.0f;
#pragma unroll
    for (int r = 0; r < 8; ++r) {
        float2 ai = attr[b * N_ + rowBase + r + 8 * half];   // uniform over lm per half
        float pd2 = ai.x + aj.x - 2.0f * c[r];
        pd2 = fminf(fmaxf(pd2, 0.0f), 50.0f);
        // pair mask, branchless: both cp (key >= 0) and different slice ids
        float m = (ai.y >= 0.0f && aj.y >= 0.0f && ai.y != aj.y) ? 1.0f : 0.0f;
        accv += m * __expf(-pd2);
    }
    accv *= scale;
    // wave32 reduction
    for (int o = 16; o > 0; o >>= 1) accv += __shfl_xor(accv, o, 32);
    if (lane == 0) atomicAdd(&rep[b], accv);
}

// ---------------- K4: finalize 4 scalars ----------------
__global__ void k_final(const float* seg, const float* rep, const float* acc,
                        const int* hist, const int* cnt, const int* ncp,
                        float* out) {
    if (threadIdx.x != 0 || blockIdx.x != 0) return;
    float sum_valid = 0.f, tot = 0.f, sbl = 0.f, sat = 0.f, srp = 0.f;
    for (int b = 0; b < B_; ++b) {
        float fncp  = (float)ncp[b];
        float valid = (fncp > 0.0f) ? 1.0f : 0.0f;
        float fnn   = (float)N_ - fncp;

        float bce = acc[b * 4 + 0] / (float)N_;
        float pm  = acc[b * 4 + 1] / fmaxf(fncp, 1.0f);
        float nm  = acc[b * 4 + 2] / fmaxf(fnn, 1.0f);
        float sp  = acc[b * 4 + 3] / fmaxf(fnn, 1.0f);
        float bl  = bce + c_MARG_W * (pm + nm) + c_NONCP_W * sp;

        float att_sum = 0.f, att_n = 0.f, h2 = 0.f;
        for (int k = 0; k < K_; ++k) {
            int h = hist[b * K_ + k];
            if (h > 0) {
                att_sum += seg[b * K_ + k] / fmaxf((float)cnt[b * K_ + k], 1.0f);
                att_n   += 1.0f;
            }
            h2 += (float)h * (float)h;
        }
        float at = c_ATTR_W * att_sum / fmaxf(att_n, 1.0f);
        float rep_cnt = fncp * fncp - h2;                // ordered cross-slice cp pairs
        float rp = c_REP_W * rep[b] / fmaxf(rep_cnt, 1.0f);

        sum_valid += valid;
        tot += valid * (bl + at + rp);
        sbl += valid * bl; sat += valid * at; srp += valid * rp;
    }
    float vb = fmaxf(sum_valid, 1.0f);
    out[0] = tot / vb;
    out[1] = sbl / vb;
    out[2] = sat / vb;
    out[3] = srp / vb;
}

extern "C" void kernel_launch(void* const* d_in, const int* in_sizes, int n_in,
                              void* d_out, int out_size, void* d_ws, size_t ws_size,
                              hipStream_t stream) {
    (void)in_sizes; (void)n_in; (void)out_size; (void)ws_size;
    const float* beta  = (const float*)d_in[0];
    const float* embed = (const float*)d_in[1];
    const int*   sid   = (const int*)d_in[2];
    const int*   cp    = (const int*)d_in[3];
    float* out = (float*)d_out;

    char* ws = (char*)d_ws;
    size_t off = 0;
    float*  wemb  = (float*)(ws + off);  off += (size_t)B_ * N_ * D_ * sizeof(float);
    float2* attr  = (float2*)(ws + off); off += (size_t)B_ * N_ * sizeof(float2);
    float*  wbeta = (float*)(ws + off);  off += (size_t)B_ * N_ * sizeof(float);
    float*  seg   = (float*)(ws + off);  off += (size_t)B_ * K_ * sizeof(float);
    float*  rep   = (float*)(ws + off);  off += (size_t)B_ * sizeof(float);
    float*  acc   = (float*)(ws + off);  off += (size_t)B_ * 4 * sizeof(float);
    int* first = (int*)(ws + off); off += (size_t)B_ * K_ * sizeof(int);
    int* hist  = (int*)(ws + off); off += (size_t)B_ * K_ * sizeof(int);
    int* cnt   = (int*)(ws + off); off += (size_t)B_ * K_ * sizeof(int);
    int* ncp   = (int*)(ws + off); off += (size_t)B_ * sizeof(int);

    k_init<<<1, 256, 0, stream>>>(seg, rep, acc, first, hist, cnt, ncp);

    k_prep<<<(B_ * N_) / 256, 256, 0, stream>>>(beta, embed, sid, cp,
                                                wemb, attr, wbeta,
                                                first, hist, cnt, ncp);

    k_point<<<(B_ * N_) / 256, 256, 0, stream>>>(sid, cp, wemb, wbeta,
                                                 first, ncp, seg, acc);

    const int T = N_ / 16;                       // 256
    const int P = T * (T + 1) / 2;               // 32896 tiles per batch
    const int totalWaves = B_ * P;               // 131584 waves
    k_rep<<<totalWaves / 8, 256, 0, stream>>>(wemb, attr, rep);

    k_final<<<1, 1, 0, stream>>>(seg, rep, acc, hist, cnt, ncp, out);
}